// PointTransformer_28063316312675
// MI455X (gfx1250) — compile-verified
//
#include <hip/hip_runtime.h>
#include <hip/hip_bf16.h>
#include <math.h>

typedef __attribute__((ext_vector_type(2))) float v2f;
typedef __attribute__((ext_vector_type(8))) float v8f;

// ---- problem constants (from reference) ----
constexpr int   Bn   = 2;     // batch
constexpr int   Ls   = 512;   // LEN_IN
constexpr int   Lt   = 513;   // LEN_OUT + 1 (start token)
constexpr int   DA   = 128;   // D_ATT
constexpr int   GC   = 16;    // N_GROUP
constexpr int   DG   = 8;     // D_GROUP
constexpr int   FFH  = 512;   // FF hidden
constexpr int   EMB  = 32;    // embedding hidden
constexpr int   LH   = 16;    // last hidden
constexpr int   DIN  = 8;
constexpr int   DOUT = 8;
constexpr float INV_SQRT_DG = 0.35355339059327373f; // 1/sqrt(8)

// ---- fp32 WMMA 16x16x4 wrapper ----
// A: 16x4 f32 -> 2 VGPRs/lane ; B: 4x16 -> 2 VGPRs/lane ; C/D: 16x16 -> 8 VGPRs/lane
__device__ __forceinline__ v8f wmma_f32_4(v2f a, v2f b, v8f c) {
    return __builtin_amdgcn_wmma_f32_16x16x4_f32(
        /*neg_a=*/false, a, /*neg_b=*/false, b,
        /*c_mod=*/(short)0, c, /*reuse_a=*/false, /*reuse_b=*/false);
}

// A fragment: lane<16 holds rows m0+lane, k-cols {k0, k0+1}; lanes 16-31 k-cols {k0+2, k0+3}
// B fragment: lane n (n=lane&15) holds cols n0+n, k-rows mirrored like A.
// C layout : lane l, elem i -> row m0 + i + (l>=16 ? 8:0), col n0 + (l&15)

// =====================================================================
// Generic (optionally z-batched) GEMM:  C = act(A[MxK] * B[KxN] + bias)
// Wave tile = 32x32 (2x2 WMMA subtiles, 4 accumulators).
// Out-of-range rows/cols are CLAMPED on load (unconditional vmem, no exec
// branching in the K-loop) and masked only at the store epilogue.
// TRANSB is a template parameter -> branch-free inner loop.
// z decomposed as (zb, zg) with zb = z/Gz, zg = z%Gz; per-part offsets.
// =====================================================================
template <int TRANSB>
__global__ __launch_bounds__(32)
void k_gemm(const float* __restrict__ A, const float* __restrict__ Bm,
            const float* __restrict__ bias, float* __restrict__ C,
            int M, int N, int K, int lda, int ldb, int ldc,
            int relu, int Gz,
            long sAb, long sAg, long sBb, long sBg, long sCb, long sCg)
{
    const int lane = threadIdx.x;
    const int lo = lane & 15;
    const int hi = lane >> 4;      // 0 or 1
    const int kof = hi * 2;        // k offset within 4-slice
    const int n0 = blockIdx.x * 32;
    const int m0 = blockIdx.y * 32;
    const int z  = blockIdx.z;
    const int zb = z / Gz, zg = z % Gz;
    A  += (size_t)zb * sAb + (size_t)zg * sAg;
    Bm += (size_t)zb * sBb + (size_t)zg * sBg;
    C  += (size_t)zb * sCb + (size_t)zg * sCg;

    // clamped (always in-bounds) load coordinates
    const int ar0 = min(m0 + lo,      M - 1);
    const int ar1 = min(m0 + 16 + lo, M - 1);
    const int bc0 = min(n0 + lo,      N - 1);
    const int bc1 = min(n0 + 16 + lo, N - 1);

    v8f acc[2][2] = {{{}, {}}, {{}, {}}};
    const int Kmain = K & ~3;

#pragma unroll 2
    for (int k0 = 0; k0 < Kmain; k0 += 4) {
        const int kk = k0 + kof;
        if (!TRANSB)   // speculative: translation faults on OOB prefetch are dropped
            __builtin_prefetch(&Bm[(size_t)(kk + 16) * ldb + bc0], 0, 1);
        v2f a0 = { A[(size_t)ar0 * lda + kk], A[(size_t)ar0 * lda + kk + 1] };
        v2f a1 = { A[(size_t)ar1 * lda + kk], A[(size_t)ar1 * lda + kk + 1] };
        v2f b0, b1;
        if (!TRANSB) {
            b0 = (v2f){ Bm[(size_t)kk * ldb + bc0], Bm[(size_t)(kk + 1) * ldb + bc0] };
            b1 = (v2f){ Bm[(size_t)kk * ldb + bc1], Bm[(size_t)(kk + 1) * ldb + bc1] };
        } else {
            b0 = (v2f){ Bm[(size_t)bc0 * ldb + kk], Bm[(size_t)bc0 * ldb + kk + 1] };
            b1 = (v2f){ Bm[(size_t)bc1 * ldb + kk], Bm[(size_t)bc1 * ldb + kk + 1] };
        }
        acc[0][0] = wmma_f32_4(a0, b0, acc[0][0]);
        acc[0][1] = wmma_f32_4(a0, b1, acc[0][1]);
        acc[1][0] = wmma_f32_4(a1, b0, acc[1][0]);
        acc[1][1] = wmma_f32_4(a1, b1, acc[1][1]);
    }

    if (Kmain < K) {   // guarded K-tail (e.g. K = 513)
        const int kk = Kmain + kof;
        v2f a0 = {0.f, 0.f}, a1 = {0.f, 0.f}, b0 = {0.f, 0.f}, b1 = {0.f, 0.f};
        if (kk < K) {
            a0.x = A[(size_t)ar0 * lda + kk];
            a1.x = A[(size_t)ar1 * lda + kk];
            if (!TRANSB) { b0.x = Bm[(size_t)kk * ldb + bc0]; b1.x = Bm[(size_t)kk * ldb + bc1]; }
            else         { b0.x = Bm[(size_t)bc0 * ldb + kk]; b1.x = Bm[(size_t)bc1 * ldb + kk]; }
        }
        if (kk + 1 < K) {
            a0.y = A[(size_t)ar0 * lda + kk + 1];
            a1.y = A[(size_t)ar1 * lda + kk + 1];
            if (!TRANSB) { b0.y = Bm[(size_t)(kk + 1) * ldb + bc0]; b1.y = Bm[(size_t)(kk + 1) * ldb + bc1]; }
            else         { b0.y = Bm[(size_t)bc0 * ldb + kk + 1];   b1.y = Bm[(size_t)bc1 * ldb + kk + 1]; }
        }
        acc[0][0] = wmma_f32_4(a0, b0, acc[0][0]);
        acc[0][1] = wmma_f32_4(a0, b1, acc[0][1]);
        acc[1][0] = wmma_f32_4(a1, b0, acc[1][0]);
        acc[1][1] = wmma_f32_4(a1, b1, acc[1][1]);
    }

    // epilogue: bias + optional relu, bounds-masked stores
#pragma unroll
    for (int sn = 0; sn < 2; ++sn) {
        const int col = n0 + sn * 16 + lo;
        if (col >= N) continue;
        const float bv = bias ? bias[col] : 0.f;
#pragma unroll
        for (int sm = 0; sm < 2; ++sm) {
#pragma unroll
            for (int i = 0; i < 8; ++i) {
                const int row = m0 + sm * 16 + i + hi * 8;
                if (row < M) {
                    float v = acc[sm][sn][i] + bv;
                    if (relu) v = fmaxf(v, 0.f);
                    C[(size_t)row * ldc + col] = v;
                }
            }
        }
    }
}

// =====================================================================
// Attention scores: S[z=(b*GC+g), q, k] = (Q.Kt)*inv_sqrt*mult + add (+causal)
// Q,K buffers: (Bn*L, 128); bias tensors laid out (b, g, q, k) so the
// epilogue reads are lane-contiguous (coalesced) -- this is the dominant
// bandwidth term of the whole model.
// =====================================================================
__global__ __launch_bounds__(32)
void k_scores(const float* __restrict__ Q, const float* __restrict__ Km,
              const float* __restrict__ addB, const float* __restrict__ multB,
              float* __restrict__ S, int Lq, int Lk, int causal)
{
    const int lane = threadIdx.x;
    const int lo = lane & 15;
    const int hi = lane >> 4;
    const int kof = hi * 2;
    const int k0 = blockIdx.x * 16;
    const int q0 = blockIdx.y * 16;
    const int z  = blockIdx.z;              // z = b*GC + g
    const int b = z / GC, g = z % GC;

    const float* Aq = Q  + (size_t)b * Lq * DA + g * DG;
    const float* Bk = Km + (size_t)b * Lk * DA + g * DG;

    // clamped, unconditional loads (masked at store time)
    const int qrc = min(q0 + lo, Lq - 1);
    const int kcc = min(k0 + lo, Lk - 1);

    v8f acc = {};
#pragma unroll
    for (int kk0 = 0; kk0 < DG; kk0 += 4) {     // K = 8 -> 2 WMMAs
        v2f a  = { Aq[(size_t)qrc * DA + kk0 + kof], Aq[(size_t)qrc * DA + kk0 + kof + 1] };
        v2f bf = { Bk[(size_t)kcc * DA + kk0 + kof], Bk[(size_t)kcc * DA + kk0 + kof + 1] };
        acc = wmma_f32_4(a, bf, acc);
    }
    const int kc = k0 + lo;
    if (kc < Lk) {
#pragma unroll
        for (int i = 0; i < 8; ++i) {
            int qq = q0 + i + hi * 8;
            if (qq < Lq) {
                size_t bi = ((size_t)z * Lq + qq) * Lk + kc;   // (b,g,q,k) layout
                float sv = acc[i] * INV_SQRT_DG * multB[bi] + addB[bi];
                if (causal && kc > qq) sv = -1.0e9f;
                S[((size_t)z * Lq + qq) * Lk + kc] = sv;
            }
        }
    }
}

// =====================================================================
// Pairwise position-bias MLP (3 -> 16 relu -> 16), add & mult in one pass.
// Block handles (b, one q, 16 k's). M-rows = k indices, cols = groups.
// Layer-1: 1 WMMA (K padded 3->4). h staged via LDS -> A frags. Layer-2: 4 WMMAs.
// Output is transposed through LDS and stored in (b, g, q, k) layout with
// 16-element contiguous k-runs per group (dense writes AND dense reads).
// =====================================================================
__global__ __launch_bounds__(32)
void k_bias_mlp(const float* __restrict__ ca, const float* __restrict__ cb,
                int Lq, int Lk,
                const float* __restrict__ Wa1, const float* __restrict__ ba1,
                const float* __restrict__ Wa2, const float* __restrict__ ba2,
                const float* __restrict__ Wm1, const float* __restrict__ bm1,
                const float* __restrict__ Wm2, const float* __restrict__ bm2,
                float* __restrict__ addO, float* __restrict__ multO)
{
    __shared__ float lds[16 * GC];
    const int lane = threadIdx.x;
    const int lo = lane & 15;
    const int hi = lane >> 4;
    const int kof = hi * 2;
    const int k0 = blockIdx.x * 16;
    const int qi = blockIdx.y;
    const int b  = blockIdx.z;

    // d[row=k_local][c] = ca[b,qi,c] - cb[b,k,c], c in {0,1,2}, padded to 4
    const int krow = min(k0 + lo, Lk - 1);
    float d0 = 0.f, d1 = 0.f;
    {
        int c0 = kof, c1 = kof + 1;
        float a0 = (c0 < 3) ? ca[((size_t)b * Lq + qi) * 3 + c0] : 0.f;
        float a1 = (c1 < 3) ? ca[((size_t)b * Lq + qi) * 3 + c1] : 0.f;
        float b0 = (c0 < 3) ? cb[((size_t)b * Lk + krow) * 3 + c0] : 0.f;
        float b1 = (c1 < 3) ? cb[((size_t)b * Lk + krow) * 3 + c1] : 0.f;
        d0 = a0 - b0;
        d1 = a1 - b1;
    }

    for (int p = 0; p < 2; ++p) {
        const float* W1 = p ? Wm1 : Wa1;
        const float* B1 = p ? bm1 : ba1;
        const float* W2 = p ? Wm2 : Wa2;
        const float* B2 = p ? bm2 : ba2;
        float* out = p ? multO : addO;

        // layer 1: h = relu(d @ W1 + b1)   (K = 3, one WMMA)
        v2f a1f = {d0, d1};
        v2f b1f = {0.f, 0.f};
        if (kof     < 3) b1f.x = W1[kof * GC + lo];
        if (kof + 1 < 3) b1f.y = W1[(kof + 1) * GC + lo];
        v8f h = {};
        h = wmma_f32_4(a1f, b1f, h);
#pragma unroll
        for (int i = 0; i < 8; ++i) {
            int rr = i + hi * 8;
            lds[rr * GC + lo] = fmaxf(h[i] + B1[lo], 0.f);   // lds[k_local][hidden]
        }
        __syncthreads();

        // layer 2: out = h @ W2 + b2   (K = 16 -> 4 WMMAs, A frags from LDS)
        v8f o = {};
#pragma unroll
        for (int t = 0; t < GC; t += 4) {
            v2f a2 = { lds[lo * GC + t + kof], lds[lo * GC + t + kof + 1] };
            v2f b2f = { W2[(t + kof) * GC + lo], W2[(t + kof + 1) * GC + lo] };
            o = wmma_f32_4(a2, b2f, o);
        }
        __syncthreads();

        // transpose o (k_local x group) -> lds[group][k_local]
#pragma unroll
        for (int i = 0; i < 8; ++i)
            lds[lo * 16 + i + hi * 8] = o[i] + B2[lo];
        __syncthreads();

        // coalesced store: for each group, a contiguous 16-wide k-run
#pragma unroll
        for (int i = 0; i < 8; ++i) {
            int g = 2 * i + hi;
            int kk = k0 + lo;
            if (kk < Lk)
                out[(((size_t)b * GC + g) * Lq + qi) * Lk + kk] = lds[g * 16 + lo];
        }
        __syncthreads();
    }
}

// ================= softmax over k (one wave per row) ==================
__global__ __launch_bounds__(32)
void k_softmax(float* __restrict__ S, int Lk)
{
    const size_t row = blockIdx.x;
    float* r = S + row * (size_t)Lk;
    const int lane = threadIdx.x;
    float mx = -3.4e38f;
    for (int i = lane; i < Lk; i += 32) mx = fmaxf(mx, r[i]);
    for (int off = 16; off > 0; off >>= 1) mx = fmaxf(mx, __shfl_xor(mx, off, 32));
    float sum = 0.f;
    for (int i = lane; i < Lk; i += 32) { float e = __expf(r[i] - mx); r[i] = e; sum += e; }
    for (int off = 16; off > 0; off >>= 1) sum += __shfl_xor(sum, off, 32);
    const float inv = 1.f / sum;
    for (int i = lane; i < Lk; i += 32) r[i] *= inv;
}

// ============ residual + LayerNorm: dst = LN(x + r)*g + b =============
__global__ __launch_bounds__(32)
void k_ln(const float* x, const float* r,
          const float* __restrict__ gam, const float* __restrict__ bet, float* dst)
{
    const size_t row = blockIdx.x;
    const int lane = threadIdx.x;
    float e[4]; float s = 0.f, ss = 0.f;
#pragma unroll
    for (int j = 0; j < 4; ++j) {
        int c = lane + j * 32;
        float v = x[row * DA + c] + r[row * DA + c];
        e[j] = v; s += v; ss += v * v;
    }
    for (int off = 16; off > 0; off >>= 1) {
        s  += __shfl_xor(s, off, 32);
        ss += __shfl_xor(ss, off, 32);
    }
    const float mean = s * (1.f / DA);
    const float var  = ss * (1.f / DA) - mean * mean;
    const float inv  = rsqrtf(var + 1e-5f);
#pragma unroll
    for (int j = 0; j < 4; ++j) {
        int c = lane + j * 32;
        dst[row * DA + c] = (e[j] - mean) * inv * gam[c] + bet[c];
    }
}

// ============ prep: trg_tok = [start; target], coord gathers ==========
__global__ __launch_bounds__(256)
void k_prep(const float* __restrict__ src, const float* __restrict__ tgt,
            const float* __restrict__ start, float* __restrict__ trg_tok,
            float* __restrict__ cs, float* __restrict__ ct)
{
    const int tid = blockIdx.x * blockDim.x + threadIdx.x;
    if (tid < Bn * Lt * DOUT) {
        int c = tid % DOUT; int l = (tid / DOUT) % Lt; int b = tid / (Lt * DOUT);
        trg_tok[tid] = (l == 0) ? start[c] : tgt[((size_t)b * Ls + (l - 1)) * DOUT + c];
    }
    if (tid < Bn * Lt * 3) {
        int c = tid % 3; int l = (tid / 3) % Lt; int b = tid / (3 * Lt);
        int ci = (c == 0) ? 0 : (c + 1);   // COORD_IDX = {0,2,3}
        ct[tid] = (l == 0) ? start[ci] : tgt[((size_t)b * Ls + (l - 1)) * DOUT + ci];
    }
    if (tid < Bn * Ls * 3) {
        int c = tid % 3; int l = (tid / 3) % Ls; int b = tid / (3 * Ls);
        int ci = (c == 0) ? 0 : (c + 1);
        cs[tid] = src[((size_t)b * Ls + l) * DIN + ci];
    }
}

// =====================================================================
// Host orchestration.
// Input leaf order (jax pytree flatten: sorted dict keys, lists/tuples in order):
//  dec[3]{ca(Wk,Wo,Wq,Wv,bk,bo,bq,bv) ff(W1,b1,W2,b2) ln1 ln2 ln3 sa(8)} = 3*26
//  dec_emb(4)=78  enc[3]{attn(8) ff(4) ln1 ln2}=82+16l  enc_emb(4)=130
//  last(4)=134  pos_add(4)=138  pos_mult(4)=142  start_token=146
//  source=147  target=148
// =====================================================================
extern "C" void kernel_launch(void* const* d_in, const int* in_sizes, int n_in,
                              void* d_out, int out_size, void* d_ws, size_t ws_size,
                              hipStream_t stream)
{
    (void)in_sizes; (void)n_in; (void)out_size; (void)ws_size;
    auto FPi = [&](int i) { return (const float*)d_in[i]; };

    const float* src   = FPi(147);
    const float* tgt   = FPi(148);
    const float* start = FPi(146);

    // ---- workspace carve ----
    float* w = (float*)d_ws;
    size_t off = 0;
    auto alloc = [&](size_t n) { float* p = w + off; off += (n + 63) & ~(size_t)63; return p; };
    float* trg_tok = alloc((size_t)Bn * Lt * DOUT);
    float* cs      = alloc((size_t)Bn * Ls * 3);
    float* ct      = alloc((size_t)Bn * Lt * 3);
    float* add_ss  = alloc((size_t)Bn * Ls * Ls * GC);
    float* mult_ss = alloc((size_t)Bn * Ls * Ls * GC);
    float* add_tt  = alloc((size_t)Bn * Lt * Lt * GC);
    float* mult_tt = alloc((size_t)Bn * Lt * Lt * GC);
    float* add_ts  = alloc((size_t)Bn * Lt * Ls * GC);
    float* mult_ts = alloc((size_t)Bn * Lt * Ls * GC);
    float* xs      = alloc((size_t)Bn * Ls * DA);
    float* xt      = alloc((size_t)Bn * Lt * DA);
    float* qb      = alloc((size_t)Bn * Lt * DA);
    float* kbuf    = alloc((size_t)Bn * Lt * DA);
    float* vb      = alloc((size_t)Bn * Lt * DA);
    float* Sb      = alloc((size_t)Bn * GC * Lt * Lt);
    float* attb    = alloc((size_t)Bn * Lt * DA);
    float* projb   = alloc((size_t)Bn * Lt * DA);
    float* hb      = alloc((size_t)Bn * Lt * FFH);

    auto gemm = [&](const float* A, const float* W, const float* bias, float* C,
                    int M, int N, int K, int lda, int ldb, int ldc, int relu) {
        dim3 g((N + 31) / 32, (M + 31) / 32, 1);
        k_gemm<0><<<g, 32, 0, stream>>>(A, W, bias, C, M, N, K, lda, ldb, ldc,
                                        relu, 1, 0, 0, 0, 0, 0, 0);
    };

    auto lnr = [&](float* x, const float* r, int lnbase, int rows) {
        k_ln<<<dim3((unsigned)rows), 32, 0, stream>>>(x, r, FPi(lnbase), FPi(lnbase + 1), x);
    };

    auto ffn = [&](float* x, int ffbase, int rows) {
        gemm(x,  FPi(ffbase + 0), FPi(ffbase + 1), hb,    rows, FFH, DA,  DA,  FFH, FFH, 1);
        gemm(hb, FPi(ffbase + 2), FPi(ffbase + 3), projb, rows, DA,  FFH, FFH, DA,  DA,  0);
    };

    auto attention = [&](const float* xq, int Lq, const float* xkv, int Lk,
                         int wbase, const float* addB, const float* multB, int causal) {
        const float* Wk = FPi(wbase + 0); const float* Wo = FPi(wbase + 1);
        const float* Wq = FPi(wbase + 2); const float* Wv = FPi(wbase + 3);
        const float* bk = FPi(wbase + 4); const float* bo = FPi(wbase + 5);
        const float* bq = FPi(wbase + 6); const float* bv = FPi(wbase + 7);
        gemm(xq,  Wq, bq, qb,   Bn * Lq, DA, DA, DA, DA, DA, 0);
        gemm(xkv, Wk, bk, kbuf, Bn * Lk, DA, DA, DA, DA, DA, 0);
        gemm(xkv, Wv, bv, vb,   Bn * Lk, DA, DA, DA, DA, DA, 0);
        dim3 gs((Lk + 15) / 16, (Lq + 15) / 16, Bn * GC);
        k_scores<<<gs, 32, 0, stream>>>(qb, kbuf, addB, multB, Sb, Lq, Lk, causal);
        k_softmax<<<dim3((unsigned)(Bn * GC * Lq)), 32, 0, stream>>>(Sb, Lk);
        // batched P @ V : per (b,g)  M=Lq, N=8, K=Lk
        dim3 ga(1, (Lq + 31) / 32, Bn * GC);
        k_gemm<0><<<ga, 32, 0, stream>>>(Sb, vb, (const float*)nullptr, attb,
                                         Lq, DG, Lk, Lk, DA, DA, 0, GC,
                                         (long)GC * Lq * Lk, (long)Lq * Lk,
                                         (long)Lk * DA, (long)DG,
                                         (long)Lq * DA, (long)DG);
        gemm(attb, Wo, bo, projb, Bn * Lq, DA, DA, DA, DA, DA, 0);
    };

    // 1) prep: trg_tok, coordinate gathers
    {
        int nthr = Bn * Lt * DOUT;
        k_prep<<<(nthr + 255) / 256, 256, 0, stream>>>(src, tgt, start, trg_tok, cs, ct);
    }

    // 2) pairwise position biases (add & mult fused), (b,g,q,k) layout
    auto biasmlp = [&](const float* caP, int Lq, const float* cbP, int Lk,
                       float* aO, float* mO) {
        dim3 g((Lk + 15) / 16, (unsigned)Lq, Bn);
        k_bias_mlp<<<g, 32, 0, stream>>>(caP, cbP, Lq, Lk,
            FPi(138), FPi(139), FPi(140), FPi(141),   // pos_add
            FPi(142), FPi(143), FPi(144), FPi(145),   // pos_mult
            aO, mO);
    };
    biasmlp(cs, Ls, cs, Ls, add_ss, mult_ss);
    biasmlp(ct, Lt, ct, Lt, add_tt, mult_tt);
    biasmlp(ct, Lt, cs, Ls, add_ts, mult_ts);

    // 3) embeddings
    gemm(src,     FPi(130), FPi(131), hb, Bn * Ls, EMB, DIN,  DIN,  EMB, EMB, 1);
    gemm(hb,      FPi(132), FPi(133), xs, Bn * Ls, DA,  EMB,  EMB,  DA,  DA,  0);
    gemm(trg_tok, FPi(78),  FPi(79),  hb, Bn * Lt, EMB, DOUT, DOUT, EMB, EMB, 1);
    gemm(hb,      FPi(80),  FPi(81),  xt, Bn * Lt, DA,  EMB,  EMB,  DA,  DA,  0);

    // 4) encoder
    for (int l = 0; l < 3; ++l) {
        int base = 82 + 16 * l;
        attention(xs, Ls, xs, Ls, base + 0, add_ss, mult_ss, 0);
        lnr(xs, projb, base + 12, Bn * Ls);
        ffn(xs, base + 8, Bn * Ls);
        lnr(xs, projb, base + 14, Bn * Ls);
    }

    // 5) decoder
    for (int l = 0; l < 3; ++l) {
        int base = 26 * l;
        attention(xt, Lt, xt, Lt, base + 18 /*sa*/, add_tt, mult_tt, 1);
        lnr(xt, projb, base + 12, Bn * Lt);
        attention(xt, Lt, xs, Ls, base + 0 /*ca*/, add_ts, mult_ts, 0);
        lnr(xt, projb, base + 14, Bn * Lt);
        ffn(xt, base + 8, Bn * Lt);
        lnr(xt, projb, base + 16, Bn * Lt);
    }

    // 6) output head
    gemm(xt, FPi(134), FPi(135), hb,             Bn * Lt, LH,   DA, DA, LH,   LH,   1);
    gemm(hb, FPi(136), FPi(137), (float*)d_out,  Bn * Lt, DOUT, LH, LH, DOUT, DOUT, 0);
}